// SpinHamiltonian_22539988370203
// MI455X (gfx1250) — compile-verified
//
#include <hip/hip_runtime.h>

// XY-model Hamiltonian on MI455X (gfx1250, wave32).
// One workgroup per sample (512 blocks x 256 threads = 8 wave32s).
// The shift table is the fixed roll(-1) stencil of a 256x256 torus, so
// neighbors are computed arithmetically: down = +256 (mod 65536, pow2 mask),
// right = next element in the quad (+ one scalar load at the quad edge).
// Each thread handles 64 float4 quads (256 sites): 2x global_load_b128 +
// 1x b32 per 4 sites, no indirection. cos via hardware v_cos_f32 (arg
// pre-scaled by 1/2pi). Accumulation via V_WMMA_F32_16X16X4_F32 with an
// all-ones B (64 f32 adds/instr on the matrix pipe, co-executes with TRANS);
// wave finish via ds_swizzle SWAPX16; cross-wave finish via LDS.

typedef __attribute__((ext_vector_type(2))) float v2f;
typedef __attribute__((ext_vector_type(8))) float v8f;

#define LSIDE   256
#define VOL     (LSIDE * LSIDE)     // 65536 sites per sample
#define TPB     256                 // threads per block (8 wave32s)
#define QUADS_PER_THREAD 64         // 64 quads * 4 sites = 256 sites/thread
#define CHUNKS  8                   // 8 chunks x 8 quads
#define INV_2PI 0.15915494309189535f

__device__ __forceinline__ float cosr(float x) {
  return __builtin_amdgcn_cosf(x * INV_2PI);   // cos(x), hw range reduction
}

__global__ __launch_bounds__(TPB)
void xy_hamiltonian_kernel(const float* __restrict__ state,
                           float*       __restrict__ out) {
  const int    s   = blockIdx.x;
  const int    t   = threadIdx.x;
  const float* row = state + (size_t)s * VOL;

  // WMMA accumulator: 16x16 f32 D/C matrix (8 VGPRs/lane).
  v8f acc = {};
  const v2f ones = {1.0f, 1.0f};

  for (int chunk = 0; chunk < CHUNKS; ++chunk) {
    // Prefetch the start of the next chunk's stretch (wraps in-row-buffer).
    {
      const int pv = (4 * (t + (chunk + 1) * 8 * TPB)) & (VOL - 1);
      __builtin_prefetch(row + pv, 0, 0);
    }

    float s0 = 0.0f, s1 = 0.0f;
#pragma unroll
    for (int j = 0; j < 8; ++j) {
      const int k = t + (chunk * 8 + j) * TPB;      // quad id, coalesced
      const int v = k * 4;                          // site index, 4-aligned
      const float4 c = *(const float4*)(row + v);
      // roll -1 along axis 0: +LSIDE with torus wrap (power-of-two mask).
      const float4 d = *(const float4*)(row + ((v + LSIDE) & (VOL - 1)));
      // roll -1 along axis 1: neighbor of the quad's last element.
      const float  x = row[(v & ~(LSIDE - 1)) | ((v + 4) & (LSIDE - 1))];

      const float q0 = cosr(d.x - c.x) + cosr(d.y - c.y)
                     + cosr(c.y - c.x) + cosr(c.z - c.y);
      const float q1 = cosr(d.z - c.z) + cosr(d.w - c.w)
                     + cosr(c.w - c.z) + cosr(x   - c.w);
      s0 += q0;
      s1 += q1;
    }

    // A-matrix 16x4 f32: lane n<16 holds {A[n][0],A[n][1]}, lane n+16 holds
    // {A[n][2],A[n][3]}. B = ones(4x16)  =>  D[m][n] += sum_k A[m][k].
    v2f a; a.x = s0; a.y = s1;
    acc = __builtin_amdgcn_wmma_f32_16x16x4_f32(
        /*neg_a=*/false, a, /*neg_b=*/false, ones,
        /*c_mod=*/(short)0, acc, /*reuse_a=*/false, /*reuse_b=*/false);
  }

  // Every column of D is identical (row sums). Per-lane partial over M:
  //   lane n<16 : sum over M=0..7 of column n ; lane n+16 : M=8..15.
  float p = acc[0] + acc[1] + acc[2] + acc[3]
          + acc[4] + acc[5] + acc[6] + acc[7];

  // Full wave sum = p[lane] + p[lane ^ 16]  (ds_swizzle SWAPX16 = 0x401F).
  const int   pb  = __float_as_int(p);
  const int   qb  = __builtin_amdgcn_ds_swizzle(pb, 0x401F);
  const float tot = p + __int_as_float(qb);        // wave total in all lanes

  // Cross-wave reduction (8 waves) via LDS.
  __shared__ float wsum[TPB / 32];
  const int lane = t & 31;
  const int wid  = t >> 5;
  if (lane == 0) wsum[wid] = tot;
  __syncthreads();
  if (t == 0) {
    float h = 0.0f;
#pragma unroll
    for (int w = 0; w < TPB / 32; ++w) h += wsum[w];
    out[s] = -h;   // BETA = 1.0
  }
}

extern "C" void kernel_launch(void* const* d_in, const int* in_sizes, int n_in,
                              void* d_out, int out_size, void* d_ws, size_t ws_size,
                              hipStream_t stream) {
  (void)n_in; (void)d_ws; (void)ws_size; (void)out_size;
  const float* state = (const float*)d_in[0];
  float*       out   = (float*)d_out;
  const int samples = in_sizes[0] / VOL;   // 512
  hipLaunchKernelGGL(xy_hamiltonian_kernel, dim3(samples), dim3(TPB), 0, stream,
                     state, out);
}